// InterpolationModel_37039797961073
// MI455X (gfx1250) — compile-verified
//
#include <hip/hip_runtime.h>

// Problem geometry (matches reference): B=256, T=2048, A=22, D=2 -> 44 floats per (b,t).
#define B_     256
#define T_     2048
#define ROW4   11                 // 44 floats = 11 float4 per timestep (176 B, 16B-aligned)
#define PERB4  (T_ * ROW4)        // float4 per batch = 22528
#define SPLIT  4                  // blocks per batch in pass 1
#define SEG4   (PERB4 / SPLIT)    // 5632 float4 per pass-1 block

typedef float vfloat4 __attribute__((ext_vector_type(4)));

// ---------------------------------------------------------------------------
// Pass 0: seed workspace sentinels (ws is poisoned by the harness).
// g_first[b] = T (min-identity), g_last[b] = -1 (max-identity).
// ---------------------------------------------------------------------------
__global__ __launch_bounds__(256) void init_kernel(int* __restrict__ g_first,
                                                   int* __restrict__ g_last) {
  const int b = threadIdx.x;
  g_first[b] = T_;
  g_last[b]  = -1;
}

// ---------------------------------------------------------------------------
// Pass 1: locate the NaN gap. 4 blocks per batch (grid = (SPLIT, B)), 256
// threads (8 wave32s). Fully coalesced b128 streaming scan; per-thread
// min/max NaN timestep; LDS tree reduction; one atomicMin/atomicMax per
// block (commutative ops -> deterministic result).
// ---------------------------------------------------------------------------
__global__ __launch_bounds__(256) void find_gap_kernel(
    const vfloat4* __restrict__ x4,
    int* __restrict__ g_first, int* __restrict__ g_last) {
  __shared__ int smin[256];
  __shared__ int smax[256];

  const int b    = blockIdx.y;
  const int tid  = threadIdx.x;
  const int base = blockIdx.x * SEG4;
  const vfloat4* p = x4 + (size_t)b * PERB4;

  int tmin = T_;   // sentinel: no NaN seen
  int tmax = -1;

  for (int i = base + tid; i < base + SEG4; i += 256) {
    vfloat4 v = p[i];
    // gfx1250 speculative prefetch (global_prefetch_b8), unconditional: OOB
    // speculative prefetches are silently dropped, no exec-mask branch needed.
    __builtin_prefetch(p + i + 2048, 0, 1);
    const bool anynan = (v.x != v.x) | (v.y != v.y) | (v.z != v.z) | (v.w != v.w);
    if (anynan) {
      const int t = i / ROW4;            // constant-divide -> mul/shift
      tmin = min(tmin, t);
      tmax = max(tmax, t);
    }
  }

  smin[tid] = tmin;
  smax[tid] = tmax;
  __syncthreads();
  for (int s = 128; s > 0; s >>= 1) {
    if (tid < s) {
      smin[tid] = min(smin[tid], smin[tid + s]);
      smax[tid] = max(smax[tid], smax[tid + s]);
    }
    __syncthreads();
  }

  if (tid == 0 && smax[0] >= 0) {
    atomicMin(&g_first[b], smin[0]);
    atomicMax(&g_last[b],  smax[0]);
  }
}

// ---------------------------------------------------------------------------
// Pass 2: streaming apply. grid = (PERB4/256, B); one float4 per thread,
// coalesced b128 load + nontemporal b128 store (keeps x resident in the
// 192 MB L2 instead of letting the write stream evict it). in_gap is
// naturally false for gapless batches (fn=T, ln=-1). Rounding matches the
// reference exactly: separate RN mul/add, no FMA contraction.
// ---------------------------------------------------------------------------
__global__ __launch_bounds__(256) void apply_kernel(
    const vfloat4* __restrict__ x4, vfloat4* __restrict__ out4,
    const int* __restrict__ g_first, const int* __restrict__ g_last) {
  const int b   = blockIdx.y;
  const int i   = blockIdx.x * 256 + threadIdx.x;   // 0 .. PERB4-1
  const int t   = i / ROW4;
  const int c   = i - t * ROW4;

  const int fn = g_first[b];        // first NaN timestep (T if none)
  const int ln = g_last[b];         // last  NaN timestep (-1 if none)

  const vfloat4* pb = x4 + (size_t)b * PERB4;
  vfloat4 v;
  if (t >= fn && t <= ln) {
    const int start = fn - 1;
    const int end   = ln + 1;
    const vfloat4 xs = pb[start * ROW4 + c];   // x[b, start] row chunk (L2/L0 hot)
    const vfloat4 xe = pb[end   * ROW4 + c];   // x[b, end]   row chunk
    const float w    = (float)(t - start) / (float)(end - start);
    const float omw  = __fsub_rn(1.0f, w);
    v.x = __fadd_rn(__fmul_rn(xs.x, omw), __fmul_rn(xe.x, w));
    v.y = __fadd_rn(__fmul_rn(xs.y, omw), __fmul_rn(xe.y, w));
    v.z = __fadd_rn(__fmul_rn(xs.z, omw), __fmul_rn(xe.z, w));
    v.w = __fadd_rn(__fmul_rn(xs.w, omw), __fmul_rn(xe.w, w));
  } else {
    v = pb[i];
  }
  __builtin_nontemporal_store(v, &out4[(size_t)b * PERB4 + i]);
}

// ---------------------------------------------------------------------------
extern "C" void kernel_launch(void* const* d_in, const int* in_sizes, int n_in,
                              void* d_out, int out_size, void* d_ws, size_t ws_size,
                              hipStream_t stream) {
  const vfloat4* x4   = (const vfloat4*)d_in[0];  // float32 input; rows are 176 B -> 16B aligned
  vfloat4*       out4 = (vfloat4*)d_out;
  int* g_first = (int*)d_ws;                      // [256]
  int* g_last  = g_first + B_;                    // [256]  (2 KB total scratch)

  init_kernel<<<1, 256, 0, stream>>>(g_first, g_last);

  dim3 grid1(SPLIT, B_);
  find_gap_kernel<<<grid1, 256, 0, stream>>>(x4, g_first, g_last);

  dim3 grid2(PERB4 / 256, B_);                    // (88, 256)
  apply_kernel<<<grid2, 256, 0, stream>>>(x4, out4, g_first, g_last);
}